// EdgeEncoding_4157528343276
// MI455X (gfx1250) — compile-verified
//
#include <hip/hip_runtime.h>

typedef __attribute__((ext_vector_type(2))) float v2f;
typedef __attribute__((ext_vector_type(4))) float v4f;
typedef __attribute__((ext_vector_type(8))) float v8f;

// ---------------------------------------------------------------------------
// Kernel 1: zero-fill the [N,N] output (64 MB). float4 stores, fully coalesced.
// ---------------------------------------------------------------------------
__global__ void ee_zero_kernel(float* __restrict__ out, long long n) {
    long long i = ((long long)blockIdx.x * blockDim.x + threadIdx.x) * 4;
    if (i + 4 <= n) {
        v4f z = {0.f, 0.f, 0.f, 0.f};
        *reinterpret_cast<v4f*>(out + i) = z;
    } else {
        for (long long j = i; j < n; ++j) out[j] = 0.f;
    }
}

// ---------------------------------------------------------------------------
// Kernel 2: per-pair masked path dots via V_WMMA_F32_16X16X4_F32.
//
// One wave32 handles 16 pairs (the 16 A-matrix rows).  For hop l and K-chunk c:
//   A[m][k] = mask(p_m,l) * edge_attr[path_idx[p_m, l]][4c + k]
//   B[k][n] = edge_vector[l][4c + k]  for all 16 columns n  (broadcast)
// => D[m][n] = sum over hops/chunks of the masked dot  (same for every n).
//
// A-fragment (32-bit A 16x4): lanes 0-15 -> K{0,1}, lanes 16-31 -> K{2,3},
// row M = lane & 15.  B-fragment mirrors the K split across half-waves.
// D-fragment: VGPR r = row r (lanes 0-15) / row r+8 (lanes 16-31); all 16
// columns identical, so lane 0 holds sums for rows 0..7, lane 16 rows 8..15.
//
// Branchless masking: padded path_idx entries are valid edge indices, so A
// loads are always performed (clamped pair index) and zeroed via one per-lane
// float mask multiply -- no exec save/restore in the hot loop.  Two
// accumulators (even/odd chunks) halve the dependent-WMMA chain; their
// elementwise sum is exact since each D element is an independent running sum.
// ---------------------------------------------------------------------------
template <int NC>  // compile-time chunk count (D/4); 0 = runtime
__global__ void ee_path_dot_kernel(
    const float* __restrict__ edge_attr,    // [E, D]
    const float* __restrict__ edge_vector,  // [L, D]
    const int*   __restrict__ path_idx,     // [P, L]
    const int*   __restrict__ path_len,     // [P]
    const int*   __restrict__ pair_id,      // [P]
    float*       __restrict__ out,          // [N*N] (pre-zeroed)
    int n_pairs, int L, int nChunksRt)
{
    const int lane = threadIdx.x & 31;
    const int wave = threadIdx.x >> 5;
    const int half = lane >> 4;          // 0 -> K{0,1}, 1 -> K{2,3}
    const int m    = lane & 15;          // A-matrix row (pair within group)
    const int nChunks = (NC > 0) ? NC : nChunksRt;
    const int D       = nChunks * 4;

    const long long base =
        ((long long)blockIdx.x * (blockDim.x >> 5) + wave) * 16;
    const long long p  = base + m;
    const long long pc = (p < (long long)n_pairs) ? p : (long long)(n_pairs - 1);
    const bool inRange = (p < (long long)n_pairs);
    const int  len     = path_len[pc];

    v8f acc0 = {0.f, 0.f, 0.f, 0.f, 0.f, 0.f, 0.f, 0.f};
    v8f acc1 = {0.f, 0.f, 0.f, 0.f, 0.f, 0.f, 0.f, 0.f};

    for (int l = 0; l < L; ++l) {
        // per-lane row mask: 1.0 iff this pair exists and hop l is valid
        const float msk = (inRange && (l < len)) ? 1.0f : 0.0f;
        const long long eidx = (long long)path_idx[pc * L + l];
        const float* __restrict__ arow =
            edge_attr + eidx * (long long)D + half * 2;
        const float* __restrict__ brow =
            edge_vector + (long long)l * D + half * 2;

#pragma unroll
        for (int c = 0; c < nChunks; ++c) {
            v2f a;
            a.x = arow[c * 4 + 0] * msk;
            a.y = arow[c * 4 + 1] * msk;
            v2f b;
            b.x = brow[c * 4 + 0];
            b.y = brow[c * 4 + 1];
            if (c & 1) {
                acc1 = __builtin_amdgcn_wmma_f32_16x16x4_f32(
                    false, a, false, b, (short)0, acc1, false, false);
            } else {
                acc0 = __builtin_amdgcn_wmma_f32_16x16x4_f32(
                    false, a, false, b, (short)0, acc0, false, false);
            }
        }
    }

    const v8f acc = acc0 + acc1;  // elementwise: exact merge of partial sums

    // Lanes 0 and 16 each own 8 finished sums: divide by count and scatter.
    if (m == 0) {
        const long long mb = base + half * 8;
#pragma unroll
        for (int i = 0; i < 8; ++i) {
            const long long q = mb + i;
            if (q < (long long)n_pairs) {
                const float s   = acc[i];
                const int   cnt = path_len[q];
                const float v   = (cnt > 0) ? (s / (float)cnt) : 0.0f;
                out[(long long)pair_id[q]] = v;
            }
        }
    }
}

extern "C" void kernel_launch(void* const* d_in, const int* in_sizes, int n_in,
                              void* d_out, int out_size, void* d_ws, size_t ws_size,
                              hipStream_t stream) {
    // setup_inputs order: x, edge_attr, edge_vector, path_idx, path_len, pair_id
    const float* edge_attr   = (const float*)d_in[1];
    const float* edge_vector = (const float*)d_in[2];
    const int*   path_idx    = (const int*)d_in[3];
    const int*   path_len    = (const int*)d_in[4];
    const int*   pair_id     = (const int*)d_in[5];
    float*       out         = (float*)d_out;

    const int n_pairs = in_sizes[4];                 // P
    const int L       = in_sizes[3] / n_pairs;       // max path length (5)
    const int D       = in_sizes[2] / L;             // edge dim (64)
    const int nChunks = D / 4;

    // 1) zero the [N,N] output
    {
        const long long n = (long long)out_size;
        const long long threads = (n + 3) / 4;
        const int blk = 256;
        const long long grid = (threads + blk - 1) / blk;
        ee_zero_kernel<<<(unsigned)grid, blk, 0, stream>>>(out, n);
    }

    // 2) WMMA gather-dot-scatter: 16 pairs per wave, 8 waves per block
    {
        const int wavesPerBlock = 8;                 // 256 threads
        const int pairsPerBlock = wavesPerBlock * 16;
        const int groups = (n_pairs + pairsPerBlock - 1) / pairsPerBlock;
        if (D == 64) {
            ee_path_dot_kernel<16><<<groups, wavesPerBlock * 32, 0, stream>>>(
                edge_attr, edge_vector, path_idx, path_len, pair_id, out,
                n_pairs, L, nChunks);
        } else {
            ee_path_dot_kernel<0><<<groups, wavesPerBlock * 32, 0, stream>>>(
                edge_attr, edge_vector, path_idx, path_len, pair_id, out,
                n_pairs, L, nChunks);
        }
    }
}